// SAGENet_74251394613509
// MI455X (gfx1250) — compile-verified
//
#include <hip/hip_runtime.h>

typedef __attribute__((ext_vector_type(2))) float v2f;
typedef __attribute__((ext_vector_type(8))) float v8f;

#define FEAT 128
#define TILE_M 16
#define LDP 132   // padded LDS row stride (floats): even (8B align for b64) + bank-spread

// --- degree count: deg[dst] += 1 (int, native global_atomic_add_u32) --------
__global__ void count_deg_kernel(const int* __restrict__ dst,
                                 int* __restrict__ deg, int E) {
  int e = blockIdx.x * blockDim.x + threadIdx.x;
  if (e < E) atomicAdd(&deg[dst[e]], 1);
}

// --- single-block exclusive scan of deg -> offs[0..N]; also inv = 1/max(d,1)
__global__ void __launch_bounds__(1024)
scan_kernel(const int* __restrict__ deg, int* __restrict__ offs,
            float* __restrict__ inv, int N) {
  __shared__ int sdata[1024];
  __shared__ int running;
  const int tid = threadIdx.x;
  if (tid == 0) running = 0;
  __syncthreads();
  for (int base = 0; base < N; base += 1024) {
    int i = base + tid;
    int v = (i < N) ? deg[i] : 0;
    sdata[tid] = v;
    __syncthreads();
    // Hillis-Steele inclusive scan
    for (int off = 1; off < 1024; off <<= 1) {
      int t = (tid >= off) ? sdata[tid - off] : 0;
      __syncthreads();
      sdata[tid] += t;
      __syncthreads();
    }
    int excl = sdata[tid] - v;
    if (i < N) {
      offs[i] = running + excl;
      inv[i] = 1.0f / fmaxf((float)v, 1.0f);
    }
    __syncthreads();
    if (tid == 1023) running += sdata[1023];
    __syncthreads();
  }
  if (tid == 0) offs[N] = running;
}

// --- cursor init -------------------------------------------------------------
__global__ void copy_offs_kernel(const int* __restrict__ offs,
                                 int* __restrict__ cursor, int N) {
  int i = blockIdx.x * blockDim.x + threadIdx.x;
  if (i < N) cursor[i] = offs[i];
}

// --- CSR fill: bucket src by dst (atomic bump) ------------------------------
__global__ void fill_csr_kernel(const int* __restrict__ src,
                                const int* __restrict__ dst,
                                int* __restrict__ cursor,
                                int* __restrict__ csr_src, int E) {
  int e = blockIdx.x * blockDim.x + threadIdx.x;
  if (e < E) {
    int pos = atomicAdd(&cursor[dst[e]], 1);
    csr_src[pos] = src[e];
  }
}

// --- gather-mean: one wave per node, 4 floats per lane, no atomics ----------
__global__ void __launch_bounds__(256)
gather_mean_kernel(const float* __restrict__ h, const int* __restrict__ csr,
                   const int* __restrict__ offs, const float* __restrict__ inv,
                   float* __restrict__ mean, int N) {
  const int wv = threadIdx.x >> 5;
  const int lane = threadIdx.x & 31;
  const int node = blockIdx.x * 8 + wv;
  if (node >= N) return;
  const int beg = offs[node];
  const int end = offs[node + 1];
  const int c = lane << 2;
  float a0 = 0.f, a1 = 0.f, a2 = 0.f, a3 = 0.f;
  int e = beg;
  for (; e + 4 <= end; e += 4) {
    int s0 = csr[e + 0], s1 = csr[e + 1], s2 = csr[e + 2], s3 = csr[e + 3];
    float4 v0 = *(const float4*)(h + (size_t)s0 * FEAT + c);
    float4 v1 = *(const float4*)(h + (size_t)s1 * FEAT + c);
    float4 v2 = *(const float4*)(h + (size_t)s2 * FEAT + c);
    float4 v3 = *(const float4*)(h + (size_t)s3 * FEAT + c);
    a0 += v0.x + v1.x + v2.x + v3.x;
    a1 += v0.y + v1.y + v2.y + v3.y;
    a2 += v0.z + v1.z + v2.z + v3.z;
    a3 += v0.w + v1.w + v2.w + v3.w;
  }
  for (; e < end; ++e) {
    int s = csr[e];
    float4 v = *(const float4*)(h + (size_t)s * FEAT + c);
    a0 += v.x; a1 += v.y; a2 += v.z; a3 += v.w;
  }
  const float iv = inv[node];
  float4 r;
  r.x = a0 * iv; r.y = a1 * iv; r.z = a2 * iv; r.w = a3 * iv;
  *(float4*)(mean + (size_t)node * FEAT + c) = r;
}

// --- fused SAGE layer GEMM: out = relu?(mean @ Wl + x @ Wr + bl) ------------
// One wave computes one 16x16 output tile via V_WMMA_F32_16X16X4_F32.
__global__ void __launch_bounds__(256)
sage_gemm_kernel(const float* __restrict__ mean, const float* __restrict__ x,
                 const float* __restrict__ Wl, const float* __restrict__ bl,
                 const float* __restrict__ Wr,
                 float* __restrict__ out, int ntiles, int relu) {
  extern __shared__ float lds[];
  float* sWl = lds;                    // W^T: sWl[n*LDP + k], 128 x LDP
  float* sWr = sWl + FEAT * LDP;
  float* sAm = sWr + FEAT * LDP;       // mean tile: [16][LDP]
  float* sAx = sAm + TILE_M * LDP;     // x tile:    [16][LDP]
  float* sB  = sAx + TILE_M * LDP;     // bias[128]

  const int tid = threadIdx.x;

  // Stage transposed weights once per block (row-major k x n -> [n][k]).
  for (int i = tid; i < (FEAT * FEAT / 4); i += 256) {
    int k  = i >> 5;            // 0..127
    int n0 = (i & 31) << 2;     // 0..124
    float4 vl = *(const float4*)(Wl + (size_t)k * FEAT + n0);
    float4 vr = *(const float4*)(Wr + (size_t)k * FEAT + n0);
    sWl[(n0 + 0) * LDP + k] = vl.x;
    sWl[(n0 + 1) * LDP + k] = vl.y;
    sWl[(n0 + 2) * LDP + k] = vl.z;
    sWl[(n0 + 3) * LDP + k] = vl.w;
    sWr[(n0 + 0) * LDP + k] = vr.x;
    sWr[(n0 + 1) * LDP + k] = vr.y;
    sWr[(n0 + 2) * LDP + k] = vr.z;
    sWr[(n0 + 3) * LDP + k] = vr.w;
  }
  if (tid < FEAT) sB[tid] = bl[tid];
  __syncthreads();

  const int lane    = tid & 31;   // wave32
  const int wv      = tid >> 5;   // 8 waves -> 8 column tiles
  const int half    = lane >> 4;
  const int l15     = lane & 15;
  const int colBase = wv * 16;

  for (int t = blockIdx.x; t < ntiles; t += gridDim.x) {
    const int nb = t * TILE_M;

    // Stage A tiles (mean and x) into LDS.
    for (int i = tid; i < (TILE_M * FEAT / 4); i += 256) {
      int r = i >> 5;
      int c = (i & 31) << 2;
      int node = nb + r;
      float4 va = *(const float4*)(mean + (size_t)node * FEAT + c);
      float4 vx = *(const float4*)(x    + (size_t)node * FEAT + c);
      *(float4*)(sAm + r * LDP + c) = va;
      *(float4*)(sAx + r * LDP + c) = vx;
    }
    __syncthreads();

    v8f acc = {0.f, 0.f, 0.f, 0.f, 0.f, 0.f, 0.f, 0.f};
    // A operand: lane<16 -> row l15, K=k0..k0+1 ; lane>=16 -> K=k0+2..k0+3
    const float* pAm = sAm + l15 * LDP + 2 * half;
    const float* pAx = sAx + l15 * LDP + 2 * half;
    // B operand from W^T: col = colBase+l15, same K striping
    const float* pWl = sWl + (colBase + l15) * LDP + 2 * half;
    const float* pWr = sWr + (colBase + l15) * LDP + 2 * half;

#pragma unroll
    for (int k0 = 0; k0 < FEAT; k0 += 4) {
      v2f am = *(const v2f*)(pAm + k0);
      v2f ax = *(const v2f*)(pAx + k0);
      v2f bw = *(const v2f*)(pWl + k0);
      v2f br = *(const v2f*)(pWr + k0);
      acc = __builtin_amdgcn_wmma_f32_16x16x4_f32(false, am, false, bw,
                                                  (short)0, acc, false, false);
      acc = __builtin_amdgcn_wmma_f32_16x16x4_f32(false, ax, false, br,
                                                  (short)0, acc, false, false);
    }

    // Epilogue: C VGPR r holds M=r (lanes 0-15) / M=r+8 (lanes 16-31), N=l15.
    float bias = sB[colBase + l15];
#pragma unroll
    for (int r = 0; r < 8; ++r) {
      int row = nb + r + 8 * half;
      float v = acc[r] + bias;
      if (relu) v = fmaxf(v, 0.0f);
      out[(size_t)row * FEAT + colBase + l15] = v;
    }
    __syncthreads();
  }
}

extern "C" void kernel_launch(void* const* d_in, const int* in_sizes, int n_in,
                              void* d_out, int out_size, void* d_ws, size_t ws_size,
                              hipStream_t stream) {
  const float* x   = (const float*)d_in[0];
  const int*   ei  = (const int*)d_in[1];
  const float* Wl0 = (const float*)d_in[2];
  const float* bl0 = (const float*)d_in[3];
  const float* Wr0 = (const float*)d_in[4];
  const float* Wl1 = (const float*)d_in[5];
  const float* bl1 = (const float*)d_in[6];
  const float* Wr1 = (const float*)d_in[7];
  const float* Wl2 = (const float*)d_in[8];
  const float* bl2 = (const float*)d_in[9];
  const float* Wr2 = (const float*)d_in[10];

  const int N = in_sizes[0] / FEAT;
  const int E = in_sizes[1] / 2;
  const int* src = ei;
  const int* dst = ei + E;

  // Workspace layout (all 4-byte elements).
  float* ws   = (float*)d_ws;
  float* mean = ws;
  float* h1   = mean + (size_t)N * FEAT;
  float* h2   = h1 + (size_t)N * FEAT;
  float* inv  = h2 + (size_t)N * FEAT;
  int*   deg    = (int*)(inv + N);
  int*   offs   = deg + N;        // N+1
  int*   cursor = offs + N + 1;
  int*   csrsrc = cursor + N;     // E

  const int ntiles = N / TILE_M;
  const int gemmGrid = ntiles < 625 ? ntiles : 625;
  const size_t ldsBytes =
      (size_t)(2 * FEAT * LDP + 2 * TILE_M * LDP + FEAT) * sizeof(float);
  const int gatherGrid = (N + 7) / 8;

  // --- Build CSR (by destination) once; reused by all 3 layers. ---
  hipMemsetAsync(deg, 0, (size_t)N * sizeof(int), stream);
  count_deg_kernel<<<(E + 255) / 256, 256, 0, stream>>>(dst, deg, E);
  scan_kernel<<<1, 1024, 0, stream>>>(deg, offs, inv, N);
  copy_offs_kernel<<<(N + 255) / 256, 256, 0, stream>>>(offs, cursor, N);
  fill_csr_kernel<<<(E + 255) / 256, 256, 0, stream>>>(src, dst, cursor,
                                                       csrsrc, E);

  // Layer 0
  gather_mean_kernel<<<gatherGrid, 256, 0, stream>>>(x, csrsrc, offs, inv,
                                                     mean, N);
  sage_gemm_kernel<<<gemmGrid, 256, ldsBytes, stream>>>(mean, x, Wl0, bl0, Wr0,
                                                        h1, ntiles, 1);
  // Layer 1
  gather_mean_kernel<<<gatherGrid, 256, 0, stream>>>(h1, csrsrc, offs, inv,
                                                     mean, N);
  sage_gemm_kernel<<<gemmGrid, 256, ldsBytes, stream>>>(mean, h1, Wl1, bl1,
                                                        Wr1, h2, ntiles, 1);
  // Layer 2 (no relu) -> d_out
  gather_mean_kernel<<<gatherGrid, 256, 0, stream>>>(h2, csrsrc, offs, inv,
                                                     mean, N);
  sage_gemm_kernel<<<gemmGrid, 256, ldsBytes, stream>>>(mean, h2, Wl2, bl2,
                                                        Wr2, (float*)d_out,
                                                        ntiles, 0);
}